// _ChannelAttentionModule_28166395527201
// MI455X (gfx1250) — compile-verified
//
#include <hip/hip_runtime.h>
#include <hip/hip_bf16.h>
#include <math.h>

typedef __attribute__((ext_vector_type(2))) float v2f;
typedef __attribute__((ext_vector_type(8))) float v8f;

#define CCH   64          // channels
#define NTOT  65536       // D*H*W = 16*64*64
#define BATCH 8
#define KSPAN 1024        // K range per workgroup in gram kernel
#define KTILE 128         // K staged in LDS per iteration
#define GPITCH 132        // 128+4 -> lane stride mod 64 == 4 (conflict-free)
#define APITCH 68         // 64+4  -> conflict-free attn reads in apply kernel
#define FPITCH 80         // feat tile pitch: 2*80 mod 64 = 32 -> half-waves on disjoint banks

// ---------------- zero-init workspace accumulator ----------------
__global__ void zero_kernel(float* __restrict__ p, int n) {
    int i = blockIdx.x * blockDim.x + threadIdx.x;
    if (i < n) p[i] = 0.0f;
}

// ---------------- phase 1: attn[b] = feat[b] @ feat[b]^T ----------------
// grid = (NTOT/KSPAN, BATCH) = (64, 8); block = 512 (16 waves)
__global__ __launch_bounds__(512) void gram_kernel(const float* __restrict__ x,
                                                   float* __restrict__ attn) {
    __shared__ float lds[CCH * GPITCH];

    const int b   = blockIdx.y;
    const int k0  = blockIdx.x * KSPAN;
    const float* feat = x + (size_t)b * CCH * NTOT;

    const int tid  = threadIdx.x;
    const int wave = tid >> 5;
    const int lane = tid & 31;
    const int hi   = lane >> 4;     // half-wave select (K offset +2 per ISA layout)
    const int l15  = lane & 15;
    const int ti   = wave >> 2;     // output row-tile (0..3)
    const int tj   = wave & 3;      // output col-tile (0..3)

    v8f acc = {};

    for (int kc = 0; kc < KSPAN; kc += KTILE) {
        __syncthreads();            // protect LDS reads of previous iteration
        // stage 64 rows x 128 k-values, coalesced b128 loads
        for (int q = tid; q < CCH * (KTILE / 4); q += 512) {
            const int row  = q >> 5;        // 32 quads per row
            const int quad = q & 31;
            const float4 v = *(const float4*)(feat + (size_t)row * NTOT + k0 + kc + quad * 4);
            *(float4*)(&lds[row * GPITCH + quad * 4]) = v;
        }
        __syncthreads();

        // A frag: rows 16*ti+m ; B frag: rows 16*tj+n (Gram: B = feat^T)
        const float* la = &lds[(ti * 16 + l15) * GPITCH + 2 * hi];
        const float* lb = &lds[(tj * 16 + l15) * GPITCH + 2 * hi];

        #pragma unroll
        for (int k = 0; k < KTILE; k += 4) {
            v2f a  = { la[k], la[k + 1] };   // K={0,1} (lanes 0-15) / K={2,3} (lanes 16-31)
            v2f bb = { lb[k], lb[k + 1] };
            acc = __builtin_amdgcn_wmma_f32_16x16x4_f32(
                false, a, false, bb, (short)0, acc, false, false);
        }
    }

    // C/D layout: VGPR r -> M = r (lanes 0-15) or M = 8+r (lanes 16-31), N = lane&15
    float* outp = attn + ((size_t)b * CCH + ti * 16 + hi * 8) * CCH + tj * 16 + l15;
    #pragma unroll
    for (int r = 0; r < 8; ++r) {
        atomicAdd(outp + (size_t)r * CCH, acc[r]);
    }
}

// ---------------- phase 2: softmax(rowmax - attn) == exp(rowmin - a)/sum ----------------
// grid = (BATCH); block = 64; one thread per row
__global__ void softmax_kernel(const float* __restrict__ attn,
                               float* __restrict__ attn_sm) {
    const int b = blockIdx.x;
    const int c = threadIdx.x;
    const float* row = attn + ((size_t)b * CCH + c) * CCH;
    float*       dst = attn_sm + ((size_t)b * CCH + c) * CCH;

    float mn = row[0];
    #pragma unroll 8
    for (int i = 1; i < CCH; ++i) mn = fminf(mn, row[i]);

    float s = 0.0f;
    #pragma unroll 8
    for (int i = 0; i < CCH; ++i) s += __expf(mn - row[i]);

    const float inv = 1.0f / s;
    #pragma unroll 8
    for (int i = 0; i < CCH; ++i) dst[i] = __expf(mn - row[i]) * inv;
}

// ---------------- phase 3: out = beta * (attn_sm @ feat) + x ----------------
// grid = (NTOT/64, BATCH); block = 512 (16 waves): wave -> 16ch x 16col tile
// feat tile (64ch x 64col) staged once in LDS: coalesced b128 global loads,
// shared by all 16 waves, and reused for the fused "+ x" residual (no 2nd
// global read of x).
__global__ __launch_bounds__(512) void apply_kernel(const float* __restrict__ x,
                                                    const float* __restrict__ attn_sm,
                                                    const float* __restrict__ beta_p,
                                                    float* __restrict__ out) {
    __shared__ float a_lds[CCH * APITCH];   // softmaxed attention, padded
    __shared__ float f_lds[CCH * FPITCH];   // feat tile: 64 channels x 64 columns

    const int b     = blockIdx.y;
    const int n0blk = blockIdx.x * 64;
    const int tid   = threadIdx.x;
    const float* feat = x + (size_t)b * CCH * NTOT;

    // cache softmaxed 64x64 attention in LDS (padded pitch)
    for (int q = tid; q < CCH * CCH; q += 512) {
        const int row = q >> 6;
        const int col = q & 63;
        a_lds[row * APITCH + col] = attn_sm[((size_t)b * CCH + row) * CCH + col];
    }
    // stage feat tile: 64 rows x 16 float4 each, fully coalesced b128 loads
    for (int q = tid; q < CCH * 16; q += 512) {
        const int row  = q >> 4;
        const int quad = q & 15;
        const float4 v = *(const float4*)(feat + (size_t)row * NTOT + n0blk + quad * 4);
        *(float4*)(&f_lds[row * FPITCH + quad * 4]) = v;
    }
    __syncthreads();

    const int wave = tid >> 5;
    const int lane = tid & 31;
    const int hi   = lane >> 4;
    const int l15  = lane & 15;
    const int cti  = wave & 3;              // channel tile 0..3
    const int nloc = (wave >> 2) * 16;      // 16-column group within tile

    const float* la = &a_lds[(cti * 16 + l15) * APITCH + 2 * hi];
    const float* fb = &f_lds[(2 * hi) * FPITCH + nloc + l15];

    v8f acc = {};
    #pragma unroll
    for (int k = 0; k < CCH; k += 4) {
        v2f a  = { la[k], la[k + 1] };                         // A: attn tile rows
        v2f bb = { fb[k * FPITCH], fb[(k + 1) * FPITCH] };     // B: feat tile rows
        acc = __builtin_amdgcn_wmma_f32_16x16x4_f32(
            false, a, false, bb, (short)0, acc, false, false);
    }

    const float beta = beta_p[0];
    const size_t o0 = ((size_t)b * CCH + cti * 16 + hi * 8) * (size_t)NTOT + n0blk + nloc + l15;
    const float* xr = &f_lds[(cti * 16 + hi * 8) * FPITCH + nloc + l15];
    #pragma unroll
    for (int r = 0; r < 8; ++r) {
        // residual x read from the staged LDS tile (channel cti*16 + hi*8 + r)
        out[o0 + (size_t)r * NTOT] = beta * acc[r] + xr[r * FPITCH];
    }
}

extern "C" void kernel_launch(void* const* d_in, const int* in_sizes, int n_in,
                              void* d_out, int out_size, void* d_ws, size_t ws_size,
                              hipStream_t stream) {
    (void)in_sizes; (void)n_in; (void)out_size; (void)ws_size;

    const float* x    = (const float*)d_in[0];
    const float* beta = (const float*)d_in[1];
    float* out        = (float*)d_out;

    float* attn_raw = (float*)d_ws;                   // BATCH*64*64 floats
    float* attn_sm  = attn_raw + BATCH * CCH * CCH;   // BATCH*64*64 floats

    const int attn_elems = BATCH * CCH * CCH;         // 32768
    zero_kernel<<<dim3((attn_elems + 255) / 256), dim3(256), 0, stream>>>(attn_raw, attn_elems);

    gram_kernel<<<dim3(NTOT / KSPAN, BATCH), dim3(512), 0, stream>>>(x, attn_raw);

    softmax_kernel<<<dim3(BATCH), dim3(CCH), 0, stream>>>(attn_raw, attn_sm);

    apply_kernel<<<dim3(NTOT / 64, BATCH), dim3(512), 0, stream>>>(x, attn_sm, beta, out);
}